// YOLOLayer_31396210934130
// MI455X (gfx1250) — compile-verified
//
#include <hip/hip_runtime.h>
#include <stdint.h>
#include <stddef.h>

// ---------------- problem constants (from reference) ----------------
#define Gd      76
#define GGd     5776            // 76*76
#define NBd     16
#define NAd     3
#define NCd     80
#define NCHd    85              // 5 + 80
#define NTGT    200
#define NCELLS  (NBd*NAd*GGd)   // 277248
#define TILEd   76              // one grid row per block; 76 tiles * 76 cells = 5776
#define NTILES  76
#define STRIDEF 8.0f            // 608 / 76
#define OUTTILE (TILEd*NCHd)    // 6460 floats = 25,840 B (contiguous in out)
#define OUT_ELEMS ((size_t)NCELLS * NCHd)   // 23,566,080 (scalar loss goes after)

typedef unsigned int v4u __attribute__((ext_vector_type(4)));
typedef int          v4i __attribute__((ext_vector_type(4)));
typedef int          v8i __attribute__((ext_vector_type(8)));

// scaled anchors: ANCHORS / stride(=8)
static __device__ __forceinline__ float anchW(int a) { return a == 0 ? 1.5f : (a == 1 ? 2.375f : 5.0f); }
static __device__ __forceinline__ float anchH(int a) { return a == 0 ? 2.0f  : (a == 1 ? 4.5f   : 3.5f); }

static __device__ __forceinline__ float sigmoidf(float v) { return 1.0f / (1.0f + __expf(-v)); }

static __device__ __forceinline__ float bcef(float p, float t) {
    p = fminf(fmaxf(p, 1e-12f), 1.0f - 1e-12f);
    return -(t * __logf(p) + (1.0f - t) * __logf(1.0f - p));
}

static __device__ __forceinline__ float waveReduce(float v) {
    #pragma unroll
    for (int m = 16; m > 0; m >>= 1) v += __shfl_xor(v, m, 32);
    return v;
}

// D# group 0: count=1 | lds_addr | 57-bit global_addr | type=2
static __device__ __forceinline__ v4u tdm_g0(uint32_t lds, uint64_t gaddr) {
    v4u g0;
    g0[0] = 1u;
    g0[1] = lds;
    g0[2] = (uint32_t)gaddr;
    g0[3] = ((uint32_t)(gaddr >> 32) & 0x01FFFFFFu) | 0x80000000u;
    return g0;
}

static __device__ __forceinline__ void tdm_load(v4u g0, v8i g1) {
    v4i gz = { 0, 0, 0, 0 };
#if defined(__clang_major__) && (__clang_major__ >= 23)
    v8i gz8 = { 0, 0, 0, 0, 0, 0, 0, 0 };
    __builtin_amdgcn_tensor_load_to_lds(g0, g1, gz, gz, gz8, 0);
#else
    __builtin_amdgcn_tensor_load_to_lds(g0, g1, gz, gz, 0);
#endif
}

static __device__ __forceinline__ void tdm_store(v4u g0, v8i g1) {
    v4i gz = { 0, 0, 0, 0 };
#if defined(__clang_major__) && (__clang_major__ >= 23)
    v8i gz8 = { 0, 0, 0, 0, 0, 0, 0, 0 };
    __builtin_amdgcn_tensor_store_from_lds(g0, g1, gz, gz, gz8, 0);
#else
    __builtin_amdgcn_tensor_store_from_lds(g0, g1, gz, gz, 0);
#endif
}

// anchor-vs-target shape IoU + argmax (matches reference _forward prologue)
static __device__ __forceinline__ int bestAnchor(float gw, float gh, float iou3[3]) {
    int bestn = 0; float best = -1.0f;
    #pragma unroll
    for (int a = 0; a < 3; ++a) {
        float w1 = anchW(a), h1 = anchH(a);
        float inter = fminf(w1, gw) * fminf(h1, gh);
        float iou = inter / (w1 * h1 + 1e-16f + gw * gh - inter);
        iou3[a] = iou;
        if (iou > best) { best = iou; bestn = a; }
    }
    return bestn;
}

// ---------------- kernel 1: init workspace ----------------
__global__ void k_init(int* __restrict__ winner, float* __restrict__ noobj, float* __restrict__ acc) {
    int i = blockIdx.x * blockDim.x + threadIdx.x;
    if (i < NCELLS) { winner[i] = -1; noobj[i] = 1.0f; }
    if (i < 8) acc[i] = 0.0f;
}

// ---------------- kernel 2: target scatter ----------------
__global__ void k_targets(const float* __restrict__ tg, int* __restrict__ winner, float* __restrict__ noobj) {
    int t = blockIdx.x * blockDim.x + threadIdx.x;
    if (t >= NTGT) return;
    int   b  = (int)tg[t * 6 + 0];
    float gx = tg[t * 6 + 2] * (float)Gd;
    float gy = tg[t * 6 + 3] * (float)Gd;
    float gw = tg[t * 6 + 4] * (float)Gd;
    float gh = tg[t * 6 + 5] * (float)Gd;
    int gi = (int)gx, gj = (int)gy;
    float iou3[3];
    int bestn = bestAnchor(gw, gh, iou3);
    #pragma unroll
    for (int a = 0; a < 3; ++a)
        if (iou3[a] > 0.5f) noobj[(b * NAd + a) * GGd + gj * Gd + gi] = 0.0f;   // 1 -> 0 only
    int cell = (b * NAd + bestn) * GGd + gj * Gd + gi;
    noobj[cell] = 0.0f;
    atomicMax(&winner[cell], t);   // dedupe duplicate cells ("last writer" scatter semantics)
}

// ---------------- kernel 3: main stream (TDM in, TDM out) ----------------
__global__ __launch_bounds__(96)
void k_main(const float* __restrict__ x, const float* __restrict__ tg,
            const int* __restrict__ winner, const float* __restrict__ noobj,
            float* __restrict__ out, float* __restrict__ acc) {
    __shared__ float inTile[NCHd * TILEd];   // [channel][cell]  25,840 B
    __shared__ float outTile[OUTTILE];       // [cell][channel]  25,840 B

    const int gj = blockIdx.x;      // 0..75 : one grid row per block
    const int a  = blockIdx.y;      // 0..2
    const int b  = blockIdx.z;      // 0..15

    // --- TDM load: x[b, a*85 .. +85, row gj] -> LDS (wave 0 only issues) ---
    if (threadIdx.x < 32) {
        uint64_t gaddr = (uint64_t)(uintptr_t)(x + ((size_t)(b * 255 + a * NCHd)) * GGd + (size_t)gj * Gd);
        v4u g0 = tdm_g0((uint32_t)(uintptr_t)&inTile[0], gaddr);
        v8i g1;
        g1[0] = 0x00020000;               // data_size = 4B
        g1[1] = (TILEd & 0xFFFF) << 16;   // tensor_dim0 = 76
        g1[2] = (NCHd  & 0xFFFF) << 16;   // tensor_dim1 = 85
        g1[3] = (TILEd & 0xFFFF) << 16;   // tile_dim0 = 76
        g1[4] = NCHd;                     // tile_dim1 = 85, tile_dim2 = 0
        g1[5] = GGd;                      // tensor_dim0_stride = 5776
        g1[6] = 0; g1[7] = 0;
        tdm_load(g0, g1);
        __builtin_amdgcn_s_wait_tensorcnt(0);
        asm volatile("" ::: "memory");    // TDM wrote LDS behind the compiler's back
    }
    __syncthreads();

    float s_bce1 = 0.0f, s_bce0 = 0.0f, s_obj = 0.0f, s_nob = 0.0f, s_cls = 0.0f;

    const int i = threadIdx.x;            // cell-in-row == gi
    if (i < TILEd) {
        const int cidx = (b * NAd + a) * GGd + gj * Gd + i;
        const int w    = winner[cidx];
        const float nob = noobj[cidx];
        const float objf = (w >= 0) ? 1.0f : 0.0f;

        const float px    = sigmoidf(inTile[0 * TILEd + i]);
        const float py    = sigmoidf(inTile[1 * TILEd + i]);
        const float pw    = inTile[2 * TILEd + i];
        const float ph    = inTile[3 * TILEd + i];
        const float pconf = sigmoidf(inTile[4 * TILEd + i]);

        float* o = &outTile[i * NCHd];
        o[0] = (px + (float)i)  * STRIDEF;
        o[1] = (py + (float)gj) * STRIDEF;
        o[2] = __expf(pw) * anchW(a) * STRIDEF;
        o[3] = __expf(ph) * anchH(a) * STRIDEF;
        o[4] = pconf;

        s_bce1 = bcef(pconf, 1.0f) * objf;
        s_bce0 = bcef(pconf, 0.0f) * nob;
        s_obj  = objf;
        s_nob  = nob;

        const int lab = (w >= 0) ? (int)tg[w * 6 + 1] : -1;
        for (int c = 0; c < NCd; ++c) {
            float p = sigmoidf(inTile[(5 + c) * TILEd + i]);
            o[5 + c] = p;
            if (w >= 0) s_cls += bcef(p, (c == lab) ? 0.9f : 0.0f);
        }
    }

    s_bce1 = waveReduce(s_bce1);
    s_bce0 = waveReduce(s_bce0);
    s_obj  = waveReduce(s_obj);
    s_nob  = waveReduce(s_nob);
    s_cls  = waveReduce(s_cls);
    if ((threadIdx.x & 31) == 0) {
        atomicAdd(&acc[0], s_bce1);
        atomicAdd(&acc[1], s_bce0);
        atomicAdd(&acc[2], s_obj);
        atomicAdd(&acc[3], s_nob);
        atomicAdd(&acc[4], s_cls);
    }

    __syncthreads();   // out-tile complete in LDS

    // --- TDM store: LDS -> contiguous out span (1-D, 6460 elems = 25,840 B) ---
    if (threadIdx.x < 32) {
        uint64_t gaddr = (uint64_t)(uintptr_t)(out + (((size_t)(b * NAd + a) * GGd) + (size_t)gj * Gd) * NCHd);
        v4u g0 = tdm_g0((uint32_t)(uintptr_t)&outTile[0], gaddr);
        v8i g1;
        g1[0] = 0x00020000;                 // data_size = 4B
        g1[1] = (OUTTILE & 0xFFFF) << 16;   // tensor_dim0 = 6460
        g1[2] = 0;                          // tensor_dim1 unused
        g1[3] = (OUTTILE & 0xFFFF) << 16;   // tile_dim0 = 6460
        g1[4] = 0;                          // tile_dim1 = 0 (1-D)
        g1[5] = 0; g1[6] = 0; g1[7] = 0;
        asm volatile("" ::: "memory");
        tdm_store(g0, g1);
        // no explicit wait: S_ENDPGM performs an implicit wait-idle; downstream
        // kernels in the stream only read x/acc, and d_out is read after stream sync.
    }
}

// ---------------- kernel 4: CIoU over winning targets ----------------
__global__ void k_ciou(const float* __restrict__ x, const float* __restrict__ tg,
                       const int* __restrict__ winner, float* __restrict__ acc) {
    int t = blockIdx.x * blockDim.x + threadIdx.x;
    if (t >= NTGT) return;
    int   b  = (int)tg[t * 6 + 0];
    float gx = tg[t * 6 + 2] * (float)Gd;
    float gy = tg[t * 6 + 3] * (float)Gd;
    float gw = tg[t * 6 + 4] * (float)Gd;
    float gh = tg[t * 6 + 5] * (float)Gd;
    int gi = (int)gx, gj = (int)gy;
    float iou3[3];
    int a = bestAnchor(gw, gh, iou3);
    int cell = (b * NAd + a) * GGd + gj * Gd + gi;
    if (winner[cell] != t) return;   // only the surviving scatter writer contributes

    const float saw = anchW(a), sah = anchH(a);
    const float* xp = x + ((size_t)(b * 255 + a * NCHd)) * GGd + (size_t)gj * Gd + gi;
    const float px = sigmoidf(xp[0]);
    const float py = sigmoidf(xp[(size_t)GGd]);
    const float pw = xp[2 * (size_t)GGd];
    const float ph = xp[3 * (size_t)GGd];

    const float bx = px + (float)gi, by = py + (float)gj;
    const float bw = __expf(pw) * saw, bh = __expf(ph) * sah;

    // _bbox_iou_xywh(pred, tb) with the +1 corner convention
    const float b1x1 = bx - bw * 0.5f, b1x2 = bx + bw * 0.5f;
    const float b1y1 = by - bh * 0.5f, b1y2 = by + bh * 0.5f;
    const float b2x1 = gx - gw * 0.5f, b2x2 = gx + gw * 0.5f;
    const float b2y1 = gy - gh * 0.5f, b2y2 = gy + gh * 0.5f;
    const float iw = fmaxf(fminf(b1x2, b2x2) - fmaxf(b1x1, b2x1) + 1.0f, 0.0f);
    const float ih = fmaxf(fminf(b1y2, b2y2) - fmaxf(b1y1, b2y1) + 1.0f, 0.0f);
    const float inter = iw * ih;
    const float a1 = (b1x2 - b1x1 + 1.0f) * (b1y2 - b1y1 + 1.0f);
    const float a2 = (b2x2 - b2x1 + 1.0f) * (b2y2 - b2y1 + 1.0f);
    const float iou = inter / (a1 + a2 - inter + 1e-16f);

    const float tx = gx - floorf(gx), ty = gy - floorf(gy);
    const float tw = __logf(gw / saw + 1e-16f);
    const float th = __logf(gh / sah + 1e-16f);

    const float xc1 = fminf(b1x1, b2x1), xc2 = fmaxf(b1x2, b2x2);
    const float yc1 = fminf(b1y1, b2y1), yc2 = fmaxf(b1y2, b2y2);
    const float cc = (xc2 - xc1) * (xc2 - xc1) + (yc2 - yc1) * (yc2 - yc1) + 1e-7f;
    const float dd = (tx - px) * (tx - px) + (ty - py) * (ty - py);
    const float rdiou = dd / cc;

    const float k = 4.0f / (3.14159265358979323846f * 3.14159265358979323846f);
    const float dat = atanf(tw / th) - atanf(pw / ph);
    const float v = k * dat * dat;
    const float S = 1.0f - iou;
    const float alpha = v / (S + v + 1e-7f);

    atomicAdd(&acc[5], 1.0f - iou + rdiou + alpha * v);
}

// ---------------- kernel 5: finalize scalar loss ----------------
__global__ void k_fin(const float* __restrict__ acc, float* __restrict__ out) {
    float n_obj   = fmaxf(acc[2], 1.0f);
    float n_noobj = fmaxf(acc[3], 1.0f);
    float loss = acc[5] / (float)NBd                 // ciou_loss
               + acc[4] / (n_obj * (float)NCd)       // loss_cls
               + 1.0f   * acc[0] / n_obj             // OBJ_SCALE * conf_obj
               + 100.0f * acc[1] / n_noobj;          // NOOBJ_SCALE * conf_noobj
    out[OUT_ELEMS] = loss;
}

// ---------------- host launcher ----------------
extern "C" void kernel_launch(void* const* d_in, const int* in_sizes, int n_in,
                              void* d_out, int out_size, void* d_ws, size_t ws_size,
                              hipStream_t stream) {
    const float* x  = (const float*)d_in[0];   // (16, 255, 76, 76) fp32
    const float* tg = (const float*)d_in[1];   // (200, 6) fp32
    float* out = (float*)d_out;                // 23,566,080 output floats + 1 scalar loss

    // workspace: winner int[NCELLS] | noobj float[NCELLS] | acc float[8]  (~2.2 MB)
    int*   winner = (int*)d_ws;
    float* noobj  = (float*)((char*)d_ws + (size_t)NCELLS * sizeof(int));
    float* acc    = noobj + NCELLS;

    k_init<<<(NCELLS + 255) / 256, 256, 0, stream>>>(winner, noobj, acc);
    k_targets<<<1, 256, 0, stream>>>(tg, winner, noobj);
    dim3 grid(NTILES, NAd, NBd);
    k_main<<<grid, 96, 0, stream>>>(x, tg, winner, noobj, out, acc);
    k_ciou<<<1, 256, 0, stream>>>(x, tg, winner, acc);
    k_fin<<<1, 1, 0, stream>>>(acc, out);
}